// APACT_2611340116214
// MI455X (gfx1250) — compile-verified
//
#include <hip/hip_runtime.h>
#include <math.h>

typedef __attribute__((ext_vector_type(2))) float v2f;
typedef __attribute__((ext_vector_type(8))) float v8f;

#define NPIX 25600   // 160*160
#define ND   64
#define NCAND 512

// ---------------- workspace layout (float units) ----------------
static constexpr size_t O_MRE  = 0;                        // 160*80
static constexpr size_t O_MIM  = O_MRE + 160*80;           // 160*80
static constexpr size_t O_E    = O_MIM + 160*80;           // 160*2 interleaved (cos,sin)
static constexpr size_t O_YW   = O_E   + 320;              // 64*80*80
static constexpr size_t O_TRE  = O_YW  + 64*80*80;         // 64*160*80
static constexpr size_t O_TIM  = O_TRE + 64*160*80;        // 64*160*80
static constexpr size_t O_YRE  = O_TIM + 64*160*80;        // 64*25600
static constexpr size_t O_YIM  = O_YRE + (size_t)ND*NPIX;  // 64*25600
static constexpr size_t O_ABSY = O_YIM + (size_t)ND*NPIX;  // 64*25600
static constexpr size_t O_SOA  = O_ABSY+ (size_t)ND*NPIX;  // 14 arrays x 25600
static constexpr size_t O_LOSS = O_SOA + 14*(size_t)NPIX;  // 512
static constexpr size_t O_BIDX = O_LOSS+ 512;              // int (8 float pad)
static constexpr size_t O_XRE  = O_BIDX+ 8;                // 25600
static constexpr size_t O_XIM  = O_XRE + NPIX;             // 25600
static constexpr size_t O_G1R  = O_XIM + NPIX;             // 160*80
static constexpr size_t O_G1I  = O_G1R + 160*80;           // 160*80

// SoA slots
#define S_PRE 0
#define S_PIM 1
#define S_QRE 2
#define S_QIM 3
#define S_A   4
#define S_B   5
#define S_C   6
#define S_S2  7
#define S_C0  8
#define S_S0  9
#define S_CD  10
#define S_SD  11
#define S_C2T 12
#define S_S2T 13

__device__ __forceinline__ v8f wmma4(v2f a, v2f b, v8f c) {
  // V_WMMA_F32_16X16X4_F32 : D = A(16x4) * B(4x16) + C(16x16), f32 all through
  return __builtin_amdgcn_wmma_f32_16x16x4_f32(false, a, false, b, (short)0, c,
                                               false, false);
}

// ---------------- kernel 1: twiddle tables + windowing ----------------
__global__ void setup_k(const float* __restrict__ patch,
                        const float* __restrict__ gw,
                        float* __restrict__ ws) {
  const int tid = blockIdx.x * blockDim.x + threadIdx.x;
  const float TWO_PI_160 = 6.283185307179586f / 160.0f;
  if (tid < 64 * 6400) {                    // yw = patch * window
    int rem = tid % 6400;
    ws[O_YW + tid] = patch[tid] * gw[rem];
  }
  if (tid < 160 * 80) {                     // M[u][a] = (-1)^u e^{-2pi i u(a+40)/160}
    int u = tid / 80, a = tid % 80;
    int t = (u * (a + 40)) % 160;
    float ang = TWO_PI_160 * (float)t;
    float sgn = (u & 1) ? -1.0f : 1.0f;
    ws[O_MRE + tid] =  sgn * cosf(ang);
    ws[O_MIM + tid] = -sgn * sinf(ang);
  }
  if (tid < 160) {                          // E[t] = e^{+2pi i t/160}
    float ang = TWO_PI_160 * (float)tid;
    ws[O_E + 2 * tid + 0] = cosf(ang);
    ws[O_E + 2 * tid + 1] = sinf(ang);
  }
}

// ---------------- kernel 2: stage-1 DFT  T = M * yw  (WMMA) ----------------
// waves: 64 d * 10 utile * 5 btile * 2 part = 6400
__global__ void fwd1_k(const float* __restrict__ ws_ro, float* __restrict__ ws) {
  int wid  = blockIdx.x * (blockDim.x >> 5) + (threadIdx.x >> 5);
  int lane = threadIdx.x & 31;
  if (wid >= 6400) return;
  int part = wid & 1;
  int bt   = (wid >> 1) % 5;
  int ut   = (wid / 10) % 10;
  int d    = wid / 100;
  const float* A = ws_ro + (part ? O_MIM : O_MRE);
  const float* B = ws_ro + O_YW + (size_t)d * 6400;
  float*       C = ws + (part ? O_TIM : O_TRE) + (size_t)d * 12800;
  int h = lane >> 4, m = lane & 15;
  int u0 = ut * 16, b0 = bt * 16;
  v8f acc = {};
  for (int k0 = 0; k0 < 80; k0 += 4) {
    v2f a, b;
    const float* ar = A + (u0 + m) * 80 + k0 + 2 * h;
    a.x = ar[0];  a.y = ar[1];
    b.x = B[(k0 + 2 * h) * 80 + b0 + m];
    b.y = B[(k0 + 2 * h + 1) * 80 + b0 + m];
    acc = wmma4(a, b, acc);
  }
  for (int r = 0; r < 8; ++r)
    C[(u0 + r + 8 * h) * 80 + b0 + m] = acc[r];
}

// ---------------- kernel 3: stage-2 DFT  Y = T * M^T  (WMMA) ----------------
// waves: 64 d * 10 ut * 10 vt * 2 part = 12800
__global__ void fwd2_k(const float* __restrict__ ws_ro, float* __restrict__ ws) {
  int wid  = blockIdx.x * (blockDim.x >> 5) + (threadIdx.x >> 5);
  int lane = threadIdx.x & 31;
  if (wid >= 12800) return;
  int part = wid & 1;
  int vt   = (wid >> 1) % 10;
  int ut   = (wid / 20) % 10;
  int d    = wid / 200;
  const float* Mre = ws_ro + O_MRE;
  const float* Mim = ws_ro + O_MIM;
  const float* Ap[2]; const float* Bp[2]; float sg[2];
  Ap[0] = ws_ro + O_TRE + (size_t)d * 12800;
  Ap[1] = ws_ro + O_TIM + (size_t)d * 12800;
  if (part == 0) { Bp[0] = Mre; Bp[1] = Mim; sg[0] = 1.f; sg[1] = -1.f; } // Yre
  else           { Bp[0] = Mim; Bp[1] = Mre; sg[0] = 1.f; sg[1] =  1.f; } // Yim
  int h = lane >> 4, m = lane & 15;
  int u0 = ut * 16, v0 = vt * 16;
  v8f acc = {};
  for (int ps = 0; ps < 2; ++ps) {
    const float* A = Ap[ps]; const float* B = Bp[ps]; float s = sg[ps];
    for (int k0 = 0; k0 < 80; k0 += 4) {
      v2f a, b;
      const float* ar = A + (u0 + m) * 80 + k0 + 2 * h;
      a.x = s * ar[0];  a.y = s * ar[1];
      const float* br = B + (v0 + m) * 80 + k0 + 2 * h;  // B[k][n] = M[n][k]
      b.x = br[0];  b.y = br[1];
      acc = wmma4(a, b, acc);
    }
  }
  float* Y = ws + (part ? O_YIM : O_YRE) + (size_t)d * NPIX;
  for (int r = 0; r < 8; ++r)
    Y[(u0 + r + 8 * h) * 160 + v0 + m] = acc[r];
}

// ---------------- kernel 4: per-pixel precompute ----------------
__global__ void pre_k(const float* __restrict__ ws_ro, float* __restrict__ ws,
                      const float* __restrict__ k_loss,
                      const float* __restrict__ theta2D,
                      const float* __restrict__ delays) {
  int p = blockIdx.x * blockDim.x + threadIdx.x;
  if (p >= NPIX) return;
  int u = p / 160, v = p % 160;
  int su = (u + 80) % 160, sv = (v + 80) % 160;   // ifftshift source index
  float th = theta2D[su * 160 + sv];
  float k  = k_loss[p];
  const float* Yre = ws_ro + O_YRE;
  const float* Yim = ws_ro + O_YIM;
  float Pre = 0, Pim = 0, Qre = 0, Qim = 0, Aq = 0, Bq = 0, Cq = 0, S2 = 0;
  for (int d = 0; d < ND; ++d) {
    float yr = Yre[(size_t)d * NPIX + p];
    float yi = Yim[(size_t)d * NPIX + p];
    float ay = sqrtf(yr * yr + yi * yi);
    ws[O_ABSY + (size_t)d * NPIX + p] = ay;
    float sd_, cd_;
    __sincosf(k * delays[d], &sd_, &cd_);
    Pre = fmaf(yr, cd_, Pre);  Pim = fmaf(yi, cd_, Pim);
    Qre = fmaf(yr, sd_, Qre);  Qim = fmaf(yi, sd_, Qim);
    Aq  = fmaf(cd_, cd_, Aq);  Bq  = fmaf(cd_, sd_, Bq);  Cq = fmaf(sd_, sd_, Cq);
    S2  = fmaf(ay, ay, S2);
  }
  S2 *= k * k;
  float s0, c0;  __sincosf(k * delays[0], &s0, &c0);
  float dlt = delays[1] - delays[0];
  float sD, cD;  __sincosf(k * dlt, &sD, &cD);
  float s2t, c2t;  __sincosf(2.0f * th, &s2t, &c2t);
  float* soa = ws + O_SOA;
  soa[S_PRE * NPIX + p] = Pre;  soa[S_PIM * NPIX + p] = Pim;
  soa[S_QRE * NPIX + p] = Qre;  soa[S_QIM * NPIX + p] = Qim;
  soa[S_A   * NPIX + p] = Aq;   soa[S_B   * NPIX + p] = Bq;
  soa[S_C   * NPIX + p] = Cq;   soa[S_S2  * NPIX + p] = S2;
  soa[S_C0  * NPIX + p] = c0;   soa[S_S0  * NPIX + p] = s0;
  soa[S_CD  * NPIX + p] = cD;   soa[S_SD  * NPIX + p] = sD;
  soa[S_C2T * NPIX + p] = c2t;  soa[S_S2T * NPIX + p] = s2t;
}

// ---------------- kernel 5: candidate search (one block per candidate) ----
__global__ void cand_k(const float* __restrict__ ws_ro, float* __restrict__ ws,
                       const float* __restrict__ k_loss,
                       const float* __restrict__ cand_params) {
  const int ci  = blockIdx.x;
  const int tid = threadIdx.x;
  const float dc = cand_params[ci * 3 + 0];
  const float xx = cand_params[ci * 3 + 1];
  const float yy = cand_params[ci * 3 + 2];
  const float* soa  = ws_ro + O_SOA;
  const float* absY = ws_ro + O_ABSY;
  float a0 = 0.f, a1 = 0.f, a2 = 0.f;
  for (int p = tid; p < NPIX; p += 256) {
    float Pre = soa[S_PRE * NPIX + p], Pim = soa[S_PIM * NPIX + p];
    float Qre = soa[S_QRE * NPIX + p], Qim = soa[S_QIM * NPIX + p];
    float Aq  = soa[S_A   * NPIX + p], Bq  = soa[S_B   * NPIX + p];
    float Cq  = soa[S_C   * NPIX + p], S2  = soa[S_S2  * NPIX + p];
    float c0  = soa[S_C0  * NPIX + p], s0  = soa[S_S0  * NPIX + p];
    float cD  = soa[S_CD  * NPIX + p], sD  = soa[S_SD  * NPIX + p];
    float c2t = soa[S_C2T * NPIX + p], s2t = soa[S_S2T * NPIX + p];
    float k = k_loss[p];
    float w = dc + xx * c2t + yy * s2t;
    float sw, cw;  __sincosf(k * w, &sw, &cw);
    float lhs = cw * cw * Aq + 2.f * cw * sw * Bq + sw * sw * Cq;
    float rre = cw * Pre + sw * Qre;
    float rim = cw * Pim + sw * Qim;
    float aX  = sqrtf(rre * rre + rim * rim) / lhs;   // |X|
    // S1 = sum_d |Y_d| * |cw*cos(k d_d) + sw*sin(k d_d)| via Givens recurrence
    float c = c0, s = s0, S1 = 0.f;
    const float* ap = absY + p;
#pragma unroll 8
    for (int d = 0; d < ND; ++d) {
      float H = cw * c + sw * s;
      S1 = fmaf(ap[(size_t)d * NPIX], fabsf(H), S1);
      float nc = fmaf(c, cD, -s * sD);
      float ns = fmaf(s, cD,  c * sD);
      c = nc;  s = ns;
    }
    float kl2 = k * k;
    a0 += S2;
    a1 = fmaf(kl2 * aX, S1, a1);
    a2 = fmaf(kl2 * aX * aX, lhs, a2);
  }
  __shared__ float sh[3 * 256];
  sh[tid] = a0;  sh[256 + tid] = a1;  sh[512 + tid] = a2;
  __syncthreads();
  for (int off = 128; off > 0; off >>= 1) {
    if (tid < off) {
      sh[tid]       += sh[tid + off];
      sh[256 + tid] += sh[256 + tid + off];
      sh[512 + tid] += sh[512 + tid + off];
    }
    __syncthreads();
  }
  if (tid == 0)
    ws[O_LOSS + ci] = (sh[0] - 2.f * sh[256] + sh[512]) * (1.0f / 1638400.0f);
}

// ---------------- kernel 6: argmin + scalar outputs ----------------
__global__ void argmin_k(const float* __restrict__ ws_ro, float* __restrict__ ws,
                         const float* __restrict__ cand_params,
                         float* __restrict__ out_tail) {
  __shared__ float sv[512];
  __shared__ int   si[512];
  int t = threadIdx.x;
  float v = ws_ro[O_LOSS + t];
  if (v != v) v = INFINITY;          // NaN never wins (matches jnp.where(loss<best))
  sv[t] = v;  si[t] = t;
  __syncthreads();
  for (int off = 256; off > 0; off >>= 1) {
    if (t < off) {
      float v2 = sv[t + off]; int i2 = si[t + off];
      if (v2 < sv[t] || (v2 == sv[t] && i2 < si[t])) { sv[t] = v2; si[t] = i2; }
    }
    __syncthreads();
  }
  if (t == 0) {
    int bi = si[0];
    ((int*)(ws + O_BIDX))[0] = bi;
    out_tail[0] = cand_params[bi * 3 + 0];
    out_tail[1] = cand_params[bi * 3 + 1];
    out_tail[2] = cand_params[bi * 3 + 2];
    out_tail[3] = sv[0];
  }
}

// ---------------- kernel 7: rebuild best X ----------------
__global__ void reconX_k(const float* __restrict__ ws_ro, float* __restrict__ ws,
                         const float* __restrict__ k_loss,
                         const float* __restrict__ cand_params) {
  int p = blockIdx.x * blockDim.x + threadIdx.x;
  if (p >= NPIX) return;
  int bi = ((const int*)(ws_ro + O_BIDX))[0];
  float dc = cand_params[bi * 3 + 0];
  float xx = cand_params[bi * 3 + 1];
  float yy = cand_params[bi * 3 + 2];
  const float* soa = ws_ro + O_SOA;
  float k  = k_loss[p];
  float w  = dc + xx * soa[S_C2T * NPIX + p] + yy * soa[S_S2T * NPIX + p];
  float sw, cw;  __sincosf(k * w, &sw, &cw);
  float lhs = cw * cw * soa[S_A * NPIX + p] + 2.f * cw * sw * soa[S_B * NPIX + p]
            + sw * sw * soa[S_C * NPIX + p];
  float rre = cw * soa[S_PRE * NPIX + p] + sw * soa[S_QRE * NPIX + p];
  float rim = cw * soa[S_PIM * NPIX + p] + sw * soa[S_QIM * NPIX + p];
  ws[O_XRE + p] = rre / lhs;
  ws[O_XIM + p] = rim / lhs;
}

// ---------------- kernel 8: inverse transform stage 1 (over v) ----------
__global__ void reconG_k(const float* __restrict__ ws_ro, float* __restrict__ ws) {
  int t = blockIdx.x * blockDim.x + threadIdx.x;
  if (t >= 160 * 80) return;
  int u = t / 80, b = t % 80;
  const float* E   = ws_ro + O_E;
  const float* Xre = ws_ro + O_XRE;
  const float* Xim = ws_ro + O_XIM;
  float gr = 0.f, gi = 0.f;
  int n = b + 120;                       // fftshift + crop fold: index (b+120)%160
  for (int v = 0; v < 160; ++v) {
    int tt = (v * n) % 160;
    float ec = E[2 * tt], es = E[2 * tt + 1];
    float xr = Xre[u * 160 + v], xi = Xim[u * 160 + v];
    gr = fmaf(xr, ec, fmaf(-xi, es, gr));
    gi = fmaf(xr, es, fmaf( xi, ec, gi));
  }
  ws[O_G1R + t] = gr;
  ws[O_G1I + t] = gi;
}

// ---------------- kernel 9: inverse transform stage 2 (over u) → best_x ---
__global__ void reconOut_k(const float* __restrict__ ws_ro, float* __restrict__ out) {
  int t = blockIdx.x * blockDim.x + threadIdx.x;
  if (t >= 6400) return;
  int a = t / 80, b = t % 80;
  const float* E   = ws_ro + O_E;
  const float* G1r = ws_ro + O_G1R;
  const float* G1i = ws_ro + O_G1I;
  float acc = 0.f;
  int m = a + 120;
  for (int u = 0; u < 160; ++u) {
    int tt = (u * m) % 160;
    acc = fmaf(G1r[u * 80 + b], E[2 * tt], fmaf(-G1i[u * 80 + b], E[2 * tt + 1], acc));
  }
  out[t] = acc * (1.0f / 25600.0f);     // real(ifft2)/N^2, cropped
}

// ---------------- launch ----------------
extern "C" void kernel_launch(void* const* d_in, const int* in_sizes, int n_in,
                              void* d_out, int out_size, void* d_ws, size_t ws_size,
                              hipStream_t stream) {
  const float* patch   = (const float*)d_in[0];   // [1,64,80,80]
  const float* gw      = (const float*)d_in[1];   // [1,80,80]
  const float* delays  = (const float*)d_in[2];   // [64]
  const float* theta2D = (const float*)d_in[4];   // [160,160]
  const float* k_loss  = (const float*)d_in[5];   // [1,1,160,160] (== ifftshift(k2D))
  const float* cand    = (const float*)d_in[6];   // [512,3]
  float* out = (float*)d_out;                     // 6400 + 3 + 1
  float* ws  = (float*)d_ws;

  setup_k   <<<1600, 256, 0, stream>>>(patch, gw, ws);
  fwd1_k    <<< 800, 256, 0, stream>>>(ws, ws);           // 6400 waves, WMMA
  fwd2_k    <<<1600, 256, 0, stream>>>(ws, ws);           // 12800 waves, WMMA
  pre_k     <<< 100, 256, 0, stream>>>(ws, ws, k_loss, theta2D, delays);
  cand_k    <<< 512, 256, 0, stream>>>(ws, ws, k_loss, cand);
  argmin_k  <<<   1, 512, 0, stream>>>(ws, ws, cand, out + 6400);
  reconX_k  <<< 100, 256, 0, stream>>>(ws, ws, k_loss, cand);
  reconG_k  <<<  50, 256, 0, stream>>>(ws, ws);
  reconOut_k<<<  25, 256, 0, stream>>>(ws, out);
}